// RNNEmbedder_47004122087542
// MI455X (gfx1250) — compile-verified
//
#include <hip/hip_runtime.h>
#include <hip/hip_bf16.h>
#include <math.h>

#define HID 1024
#define EMBD 512

typedef __attribute__((ext_vector_type(2))) float v2f;
typedef __attribute__((ext_vector_type(8))) float v8f;

// ---------------------------------------------------------------------------
// WMMA helper: D = A(16x4 f32) * B(4x16 f32) + C(16x16 f32), wave32
// ---------------------------------------------------------------------------
__device__ __forceinline__ v8f wmma_f32x4(v2f a, v2f b, v8f c) {
#if defined(__HIP_DEVICE_COMPILE__) && __has_builtin(__builtin_amdgcn_wmma_f32_16x16x4_f32)
  // 8 args: (neg_a, A, neg_b, B, c_mod, C, reuse_a, reuse_b)
  return __builtin_amdgcn_wmma_f32_16x16x4_f32(
      false, a, false, b, (short)0, c, false, false);
#else
  return c;  // host-pass / missing-builtin stub
#endif
}

// ---------------------------------------------------------------------------
// Kernel 1: ih_all[s, i] = sum_j emb[x[s], j] * W_ih[i, j] + b_ih[i]
// Fused gather + NT-GEMM.  Block = 128 threads (4 waves), 64x64 C tile,
// each wave computes a 32x32 subtile with 4 accumulators.
// ---------------------------------------------------------------------------
__global__ __launch_bounds__(128) void ih_gemm_kernel(
    const int* __restrict__ x, const float* __restrict__ emb,
    const float* __restrict__ W_ih, const float* __restrict__ b_ih,
    float* __restrict__ ih_all) {
  const int lane = threadIdx.x & 31;
  const int wave = threadIdx.x >> 5;
  const int r  = lane & 15;   // row/col within 16-tile
  const int hs = lane >> 4;   // half-select (K pair)

  const int mSub = blockIdx.x * 64 + (wave & 1) * 32;   // sequence rows
  const int nSub = blockIdx.y * 64 + (wave >> 1) * 32;  // hidden cols

  // A rows gathered through the embedding table
  const float* arow0 = emb + (size_t)x[mSub + r] * HID;
  const float* arow1 = emb + (size_t)x[mSub + 16 + r] * HID;
  // B tile: B[k][n] = W_ih[n][k]  (NT layout: read W_ih rows along K)
  const float* brow0 = W_ih + (size_t)(nSub + r) * HID;
  const float* brow1 = W_ih + (size_t)(nSub + 16 + r) * HID;

  v8f c00 = {}, c01 = {}, c10 = {}, c11 = {};
  for (int k = 0; k < HID; k += 4) {
    const int kk = k + hs * 2;
    v2f a0 = *(const v2f*)(arow0 + kk);
    v2f a1 = *(const v2f*)(arow1 + kk);
    v2f b0 = *(const v2f*)(brow0 + kk);
    v2f b1 = *(const v2f*)(brow1 + kk);
    c00 = wmma_f32x4(a0, b0, c00);
    c01 = wmma_f32x4(a0, b1, c01);
    c10 = wmma_f32x4(a1, b0, c10);
    c11 = wmma_f32x4(a1, b1, c11);
  }

  const float bias0 = b_ih[nSub + r];
  const float bias1 = b_ih[nSub + 16 + r];
  // C layout: VGPR v holds (M = v + 8*hs, N = lane&15)
#pragma unroll
  for (int v = 0; v < 8; ++v) {
    const int row0 = mSub + hs * 8 + v;
    const int row1 = row0 + 16;
    ih_all[(size_t)row0 * HID + nSub + r]      = c00[v] + bias0;
    ih_all[(size_t)row0 * HID + nSub + 16 + r] = c01[v] + bias1;
    ih_all[(size_t)row1 * HID + nSub + r]      = c10[v] + bias0;
    ih_all[(size_t)row1 * HID + nSub + 16 + r] = c11[v] + bias1;
  }
}

// ---------------------------------------------------------------------------
// Device-wide barrier (arrive-count + generation), agent scope.
// ---------------------------------------------------------------------------
__device__ __forceinline__ void grid_barrier(unsigned* cnt, unsigned* gen,
                                             unsigned nblocks) {
  __threadfence();   // make this block's global stores device-visible
  __syncthreads();
  if (threadIdx.x == 0) {
    unsigned g = __hip_atomic_load(gen, __ATOMIC_ACQUIRE,
                                   __HIP_MEMORY_SCOPE_AGENT);
    unsigned a = __hip_atomic_fetch_add(cnt, 1u, __ATOMIC_ACQ_REL,
                                        __HIP_MEMORY_SCOPE_AGENT);
    if (a == nblocks - 1) {
      __hip_atomic_store(cnt, 0u, __ATOMIC_RELAXED, __HIP_MEMORY_SCOPE_AGENT);
      __hip_atomic_fetch_add(gen, 1u, __ATOMIC_RELEASE,
                             __HIP_MEMORY_SCOPE_AGENT);
    } else {
      while (__hip_atomic_load(gen, __ATOMIC_ACQUIRE,
                               __HIP_MEMORY_SCOPE_AGENT) == g) {
        __builtin_amdgcn_s_sleep(1);
      }
    }
  }
  __syncthreads();
}

// ---------------------------------------------------------------------------
// Kernel 2: sequential scan  h_{t+1} = tanh(ih_all[t] + W_hh h_t + b_hh)
// 32 persistent blocks x 256 threads.  Block b owns rows [32b, 32b+32):
// its 128 KB slice of W_hh lives in LDS for the entire 4096-step recurrence;
// h (4 KB) is staged to LDS each step.  One grid barrier per step.
// ---------------------------------------------------------------------------
__global__ __launch_bounds__(256) void rnn_scan_kernel(
    const float* __restrict__ W_hh, const float* __restrict__ b_hh,
    const float* __restrict__ ih_all, float* __restrict__ hbuf,
    unsigned* __restrict__ bar, int nsteps) {
  extern __shared__ float lds[];
  float* wsh = lds;             // 32*1024 floats: this block's W_hh rows
  float* hsh = lds + 32 * HID;  // 1024 floats: staged h_t

  const int tid   = threadIdx.x;
  const int rows0 = blockIdx.x * 32;
  const unsigned nblocks = gridDim.x;
  unsigned* cnt = bar;
  unsigned* gen = bar + 1;

  // Load this block's contiguous 32-row slice of W_hh into LDS (once).
  for (int idx = tid; idx < 32 * HID; idx += blockDim.x)
    wsh[idx] = W_hh[(size_t)rows0 * HID + idx];
  __syncthreads();

  const int wave = tid >> 5, lane = tid & 31;

  for (int t = 0; t < nsteps; ++t) {
    const float* hc = hbuf + (t & 1) * HID;        // h_t   (read)
    float*       hn = hbuf + ((t + 1) & 1) * HID;  // h_{t+1} (write)

    for (int j = tid; j < HID; j += blockDim.x) hsh[j] = hc[j];
    __syncthreads();

#pragma unroll
    for (int q = 0; q < 4; ++q) {
      const int lr = wave * 4 + q;            // local row 0..31
      const float* wr = wsh + lr * HID;
      float sum = 0.0f;
#pragma unroll 8
      for (int j = lane; j < HID; j += 32) sum = fmaf(wr[j], hsh[j], sum);
      for (int off = 16; off > 0; off >>= 1) sum += __shfl_down(sum, off, 32);
      if (lane == 0) {
        const int i = rows0 + lr;
        hn[i] = tanhf(ih_all[(size_t)t * HID + i] + sum + b_hh[i]);
      }
    }
    grid_barrier(cnt, gen, nblocks);
  }
}

// ---------------------------------------------------------------------------
// Kernel 3: out[e] = h_final . W_fc[e,:] + b_fc[e]   (wave per row)
// ---------------------------------------------------------------------------
__global__ __launch_bounds__(256) void fc_kernel(
    const float* __restrict__ h, const float* __restrict__ W_fc,
    const float* __restrict__ b_fc, float* __restrict__ out) {
  const int wave = threadIdx.x >> 5, lane = threadIdx.x & 31;
  const int row = blockIdx.x * 8 + wave;
  const float* wr = W_fc + (size_t)row * HID;
  float sum = 0.0f;
  for (int j = lane; j < HID; j += 32) sum = fmaf(wr[j], h[j], sum);
  for (int off = 16; off > 0; off >>= 1) sum += __shfl_down(sum, off, 32);
  if (lane == 0) out[row] = sum + b_fc[row];
}

// ---------------------------------------------------------------------------
extern "C" void kernel_launch(void* const* d_in, const int* in_sizes, int n_in,
                              void* d_out, int out_size, void* d_ws,
                              size_t ws_size, hipStream_t stream) {
  const int*   x    = (const int*)d_in[0];
  const float* emb  = (const float*)d_in[1];
  const float* W_ih = (const float*)d_in[2];
  const float* b_ih = (const float*)d_in[3];
  const float* W_hh = (const float*)d_in[4];
  const float* b_hh = (const float*)d_in[5];
  const float* W_fc = (const float*)d_in[6];
  const float* b_fc = (const float*)d_in[7];
  float* out = (float*)d_out;

  const int seq = in_sizes[0];  // 4096

  // Workspace layout: [ih_all: seq*HID f32][h ping-pong: 2*HID f32][2x u32]
  float* ih_all = (float*)d_ws;
  float* hbuf   = ih_all + (size_t)seq * HID;
  unsigned* bar = (unsigned*)(hbuf + 2 * HID);

  // Zero h0 + ping buffer + barrier counters (capturable memset node).
  hipMemsetAsync(hbuf, 0, 2 * HID * sizeof(float) + 2 * sizeof(unsigned),
                 stream);

  // 1) fused gather + WMMA GEMM
  dim3 gg(seq / 64, HID / 64);
  ih_gemm_kernel<<<gg, dim3(128), 0, stream>>>(x, emb, W_ih, b_ih, ih_all);

  // 2) persistent scan (W_hh slice + h stage in LDS: (32*1024+1024)*4 B)
  const size_t lds_bytes = (size_t)(32 * HID + HID) * sizeof(float);
  rnn_scan_kernel<<<dim3(HID / 32), dim3(256), lds_bytes, stream>>>(
      W_hh, b_hh, ih_all, hbuf, bar, seq);

  // 3) output projection; final h lives in buffer (seq & 1)
  const float* hfin = hbuf + (size_t)(seq & 1) * HID;
  fc_kernel<<<dim3(EMBD / 8), dim3(256), 0, stream>>>(hfin, W_fc, b_fc, out);
}